// SelectiveSSM_84215718740528
// MI455X (gfx1250) — compile-verified
//
#include <hip/hip_runtime.h>
#include <hip/hip_bf16.h>
#include <math.h>

// Problem constants (match reference)
#define B_ 2
#define L_ 2048
#define D_ 2048
#define N_ 16
#define R_ 64
#define E_ (R_ + 2 * N_)   // 96
#define M_ (B_ * L_)       // 4096 rows

typedef __attribute__((ext_vector_type(2))) float v2f;
typedef __attribute__((ext_vector_type(8))) float v8f;

// ---------------------------------------------------------------------------
// Kernel 1: x_dbl[M, 96] = x[M, 2048] @ W_xproj^T   (W_xproj is [96, 2048])
// One wave computes a 16-row stripe across all 6 column tiles (96 cols).
// A fragment: lanes 0-15 hold rows M=0..15 at K={0,1}; lanes 16-31 at K={2,3}.
// B fragment (B[k][n] = W[n][k]): lane n reads W[n0+n][k0+2*khalf .. +1] -> b64.
// ---------------------------------------------------------------------------
__global__ __launch_bounds__(128) void ssm_xproj_kernel(
    const float* __restrict__ x,    // [M_, D_]
    const float* __restrict__ Wx,   // [E_, D_]
    float* __restrict__ xdbl)       // [M_, E_]
{
    const int lane  = threadIdx.x & 31;
    const int wid   = threadIdx.x >> 5;
    const int laneM = lane & 15;
    const int khalf = lane >> 4;           // 0: K lo pair, 1: K hi pair
    const int m0    = blockIdx.x * 64 + wid * 16;

    v8f acc[6] = {{}, {}, {}, {}, {}, {}};

    const float* aRow = x + (size_t)(m0 + laneM) * D_ + 2 * khalf;

    #pragma unroll 2
    for (int k0 = 0; k0 < D_; k0 += 4) {
        v2f a = *(const v2f*)(aRow + k0);
        #pragma unroll
        for (int e = 0; e < 6; ++e) {
            v2f b = *(const v2f*)(Wx + (size_t)(e * 16 + laneM) * D_ + k0 + 2 * khalf);
            acc[e] = __builtin_amdgcn_wmma_f32_16x16x4_f32(
                false, a, false, b, (short)0, acc[e], false, false);
        }
    }

    // C/D layout: element j of v8f -> row M = j + 8*khalf, col N = laneM
    #pragma unroll
    for (int e = 0; e < 6; ++e) {
        #pragma unroll
        for (int j = 0; j < 8; ++j) {
            int row = m0 + j + 8 * khalf;
            xdbl[(size_t)row * E_ + e * 16 + laneM] = acc[e][j];
        }
    }
}

// ---------------------------------------------------------------------------
// Kernel 2: delta[M, D] = softplus(x_dbl[:, :64] @ W_dt^T + b_dt)
// W_dt is [D_, R_]; K = 64 (fully unrolled). One wave = 16 rows x 64 cols.
// ---------------------------------------------------------------------------
__global__ __launch_bounds__(128) void ssm_delta_kernel(
    const float* __restrict__ xdbl,  // [M_, E_]; cols 0..63 = delta_raw
    const float* __restrict__ Wdt,   // [D_, R_]
    const float* __restrict__ bdt,   // [D_]
    float* __restrict__ delta)       // [M_, D_]
{
    const int lane  = threadIdx.x & 31;
    const int wid   = threadIdx.x >> 5;
    const int laneM = lane & 15;
    const int khalf = lane >> 4;
    const int w     = blockIdx.x * 4 + wid;   // 8192 waves total
    const int mtile = w >> 5;                 // 256 M-tiles
    const int dgrp  = w & 31;                 // 32 groups of 64 cols
    const int m0    = mtile * 16;
    const int d0    = dgrp * 64;

    v8f acc[4] = {{}, {}, {}, {}};
    const float* aRow = xdbl + (size_t)(m0 + laneM) * E_ + 2 * khalf;

    #pragma unroll
    for (int k0 = 0; k0 < R_; k0 += 4) {
        v2f a = *(const v2f*)(aRow + k0);
        #pragma unroll
        for (int t = 0; t < 4; ++t) {
            v2f b = *(const v2f*)(Wdt + (size_t)(d0 + t * 16 + laneM) * R_ + k0 + 2 * khalf);
            acc[t] = __builtin_amdgcn_wmma_f32_16x16x4_f32(
                false, a, false, b, (short)0, acc[t], false, false);
        }
    }

    #pragma unroll
    for (int t = 0; t < 4; ++t) {
        const int   d    = d0 + t * 16 + laneM;
        const float bias = bdt[d];
        #pragma unroll
        for (int j = 0; j < 8; ++j) {
            int   row = m0 + j + 8 * khalf;
            float pre = acc[t][j] + bias;
            // softplus with overflow guard; expf underflow -> log1p(0) = 0 is exact
            float sp  = (pre > 20.0f) ? pre : log1pf(__expf(pre));
            delta[(size_t)row * D_ + d] = sp;
        }
    }
}

// ---------------------------------------------------------------------------
// Kernel 3: selective scan. Thread = one (d, n) state cell.
// Wave32 = 2 d-channels x 16 n. Block(256) = 16 d-channels; grid = (D/16, B).
// h and A = -exp(A_log) live in registers across all L=2048 steps.
// n-reduction via __shfl_xor butterfly within 16-lane groups (ds path).
// ---------------------------------------------------------------------------
__global__ __launch_bounds__(256) void ssm_scan_kernel(
    const float* __restrict__ x,      // [B_, L_, D_]
    const float* __restrict__ xdbl,   // [M_, E_]: cols 64..79 = B, 80..95 = C
    const float* __restrict__ delta,  // [M_, D_]
    const float* __restrict__ A_log,  // [D_, N_]
    const float* __restrict__ Dp,     // [D_]
    float* __restrict__ y)            // [B_, L_, D_]
{
    const int lane  = threadIdx.x & 31;
    const int wid   = threadIdx.x >> 5;
    const int n     = lane & 15;
    const int khalf = lane >> 4;
    const int b     = blockIdx.y;
    const int d     = blockIdx.x * 16 + wid * 2 + khalf;

    const float  Adn    = -__expf(A_log[d * N_ + n]);
    const float  dscale = Dp[d];
    const size_t rowB   = (size_t)b * L_;

    float h = 0.0f;

    for (int t = 0; t < L_; ++t) {
        const size_t r = rowB + t;

        // prefetch a few timesteps ahead (L2-resident working set)
        const size_t rp = (t + 8 < L_) ? (r + 8) : r;
        __builtin_prefetch(&delta[rp * D_ + d], 0, 0);
        __builtin_prefetch(&x[rp * D_ + d], 0, 0);

        const float dl = delta[r * D_ + d];           // broadcast across 16 lanes
        const float xv = x[r * D_ + d];               // broadcast across 16 lanes
        const float Bv = xdbl[r * E_ + R_ + n];
        const float Cv = xdbl[r * E_ + R_ + N_ + n];

        const float dA = __expf(dl * Adn);
        h = dA * h + (dl * xv) * Bv;

        float p = h * Cv;
        p += __shfl_xor(p, 1, 16);
        p += __shfl_xor(p, 2, 16);
        p += __shfl_xor(p, 4, 16);
        p += __shfl_xor(p, 8, 16);

        if (n == 0) y[r * D_ + d] = p + xv * dscale;
    }
}

// ---------------------------------------------------------------------------
// Launch: GEMM1 -> GEMM2(+softplus) -> scan, all on `stream`.
// Workspace: x_dbl (1.5 MB) then delta (32 MB), both fp32.
// ---------------------------------------------------------------------------
extern "C" void kernel_launch(void* const* d_in, const int* in_sizes, int n_in,
                              void* d_out, int out_size, void* d_ws, size_t ws_size,
                              hipStream_t stream) {
    const float* x    = (const float*)d_in[0];  // [B,L,D]
    const float* Wx   = (const float*)d_in[1];  // [96, D]
    const float* Wdt  = (const float*)d_in[2];  // [D, 64]
    const float* bdt  = (const float*)d_in[3];  // [D]
    const float* Alog = (const float*)d_in[4];  // [D, N]
    const float* Dp   = (const float*)d_in[5];  // [D]
    float*       y    = (float*)d_out;          // [B,L,D]

    float* ws_xdbl  = (float*)d_ws;                       // M_*E_ floats
    float* ws_delta = ws_xdbl + (size_t)M_ * E_;          // M_*D_ floats

    // GEMM1: 64 blocks x 4 waves, each wave a 16x96 stripe
    ssm_xproj_kernel<<<dim3(M_ / 64), dim3(128), 0, stream>>>(x, Wx, ws_xdbl);

    // GEMM2: 8192 waves (256 M-tiles x 32 col-groups), 4 waves/block
    ssm_delta_kernel<<<dim3((M_ / 16) * (D_ / 64) / 4), dim3(128), 0, stream>>>(
        ws_xdbl, Wdt, bdt, ws_delta);

    // Scan: 65536 threads = all B*D*N state cells
    ssm_scan_kernel<<<dim3(D_ / 16, B_), dim3(256), 0, stream>>>(
        x, ws_xdbl, ws_delta, Alog, Dp, y);
}